// Net_52630529245348
// MI455X (gfx1250) — compile-verified
//
#include <hip/hip_runtime.h>

typedef float v2f __attribute__((ext_vector_type(2)));
typedef float v8f __attribute__((ext_vector_type(8)));

#define HH 512
#define WW 512
#define PIX (HH * WW)     // 262144
#define IMG_N 96          // B*C
#define BATCH 32

// ---------------------------------------------------------------------------
// Kernel A: separable 3-tap gaussian blur. One wave per 16x16 output tile.
// Stage 1 (VALU): horizontal blur of 18 halo rows -> T[20][16] in LDS
//                 (rows 18,19 zeroed: K padded to 20 for 16x16x4 chunks).
// Stage 2 (WMMA): O(16x16) = V(16x20) * T(20x16), V banded vertical filter,
//                 5 chained v_wmma_f32_16x16x4_f32.
// ---------------------------------------------------------------------------
__global__ __launch_bounds__(128) void canny_blur_wmma(
    const float* __restrict__ img, const float* __restrict__ gauss,
    float* __restrict__ blurred)
{
    __shared__ float T[4][20][16];
    const int lane = threadIdx.x & 31;
    const int wv   = threadIdx.x >> 5;
    const int tid  = blockIdx.x * 4 + wv;   // global tile id (96*32*32 tiles)
    const int im   = tid >> 10;             // image index 0..95
    const int tile = tid & 1023;
    const int r0 = (tile >> 5) << 4;
    const int c0 = (tile & 31) << 4;
    const float g0 = gauss[0];              // exp(-0.5)
    const float g1 = gauss[1];              // 1.0
    const float* __restrict__ src = img + im * PIX;

    // ---- stage 1: horizontal blur, rows r0-1 .. r0+16 (18 rows x 16 cols)
    for (int idx = lane; idx < 288; idx += 32) {
        int r = idx >> 4, c = idx & 15;
        int gr = r0 + r - 1;
        int gc = c0 + c;
        float v = 0.f;
        if (gr >= 0 && gr < HH) {
            const float* row = src + gr * WW;
            float xm = (gc > 0)      ? row[gc - 1] : 0.f;
            float xc = row[gc];
            float xp = (gc < WW - 1) ? row[gc + 1] : 0.f;
            v = g0 * xm + g1 * xc + g0 * xp;
        }
        T[wv][r][c] = v;
    }
    // zero K-padding rows 18,19 (avoid NaN garbage feeding WMMA)
    T[wv][18 + (lane >> 4)][lane & 15] = 0.f;
    __syncthreads();

    // ---- stage 2: vertical blur as banded matmul
    const int rc    = lane & 15;             // A row M / B col N
    const int khalf = (lane >> 4) << 1;      // K sub-split across lane halves
    float* __restrict__ dst = blurred + im * PIX;

#if __has_builtin(__builtin_amdgcn_wmma_f32_16x16x4_f32)
    v8f acc = {0.f, 0.f, 0.f, 0.f, 0.f, 0.f, 0.f, 0.f};
#pragma unroll
    for (int k = 0; k < 5; ++k) {
        const int kb = 4 * k + khalf;
        v2f a, b;
        // V[m][k]: g0 at k==m and k==m+2, g1 at k==m+1 (vertical 3-tap band)
        a.x = (kb     == rc + 1) ? g1 : ((kb     == rc || kb     == rc + 2) ? g0 : 0.f);
        a.y = (kb + 1 == rc + 1) ? g1 : ((kb + 1 == rc || kb + 1 == rc + 2) ? g0 : 0.f);
        b.x = T[wv][kb][rc];
        b.y = T[wv][kb + 1][rc];
        acc = __builtin_amdgcn_wmma_f32_16x16x4_f32(
            false, a, false, b, (short)0, acc, false, false);
    }
    // D layout: VGPR v -> row v (lanes 0-15) / row v+8 (lanes 16-31), col lane&15
    const int mhi = (lane >> 4) << 3;
#pragma unroll
    for (int v = 0; v < 8; ++v) {
        dst[(r0 + v + mhi) * WW + (c0 + rc)] = acc[v];
    }
#else
    // VALU fallback: each lane produces 8 outputs (rows mhi.. of tile)
    const int mhi = (lane >> 4) << 3;
    (void)khalf;
#pragma unroll
    for (int v = 0; v < 8; ++v) {
        int m = v + mhi;
        float o = g0 * T[wv][m][rc] + g1 * T[wv][m + 1][rc] + g0 * T[wv][m + 2][rc];
        dst[(r0 + m) * WW + (c0 + rc)] = o;
    }
#endif
}

// ---------------------------------------------------------------------------
// Kernel B: fused Sobel -> grad_mag, grad_orientation, early_threshold
// Interior fast path (99.2% of pixels): unguarded coalesced loads.
// early_threshold is never re-read -> non-temporal store (protect L2 for
// blurred/gmag/gori which ARE re-read).
// ---------------------------------------------------------------------------
__global__ __launch_bounds__(256) void canny_grad(
    const float* __restrict__ blurred,
    float* __restrict__ grad_mag, float* __restrict__ grad_ori,
    float* __restrict__ early_thr)
{
    const int gid = blockIdx.x * 256 + threadIdx.x;   // 0 .. 32*PIX-1
    const int b   = gid >> 18;
    const int pix = gid & (PIX - 1);
    const int i = pix >> 9, j = pix & 511;

    float mag = 0.f, sx = 0.f, sy = 0.f;
    if (i > 0 && i < HH - 1 && j > 0 && j < WW - 1) {
#pragma unroll
        for (int c = 0; c < 3; ++c) {
            const float* __restrict__ p = blurred + (b * 3 + c) * PIX + i * WW + j;
            float tl = p[-WW - 1], tc = p[-WW], tr = p[-WW + 1];
            float ml = p[-1],                  mr = p[1];
            float bl = p[WW - 1],  bc = p[WW], br = p[WW + 1];
            float gx = (tl - tr) + 2.f * (ml - mr) + (bl - br);
            float gy = (tl + 2.f * tc + tr) - (bl + 2.f * bc + br);
            mag += sqrtf(gx * gx + gy * gy + 1e-5f);
            sx += gx;
            sy += gy;
        }
    } else {
        const bool up = i > 0, dn = i < HH - 1, lf = j > 0, rt = j < WW - 1;
#pragma unroll
        for (int c = 0; c < 3; ++c) {
            const float* __restrict__ p = blurred + (b * 3 + c) * PIX;
            float tl = (up && lf) ? p[(i - 1) * WW + (j - 1)] : 0.f;
            float tc = up         ? p[(i - 1) * WW + j]       : 0.f;
            float tr = (up && rt) ? p[(i - 1) * WW + (j + 1)] : 0.f;
            float ml = lf         ? p[i * WW + (j - 1)]       : 0.f;
            float mr = rt         ? p[i * WW + (j + 1)]       : 0.f;
            float bl = (dn && lf) ? p[(i + 1) * WW + (j - 1)] : 0.f;
            float bc = dn         ? p[(i + 1) * WW + j]       : 0.f;
            float br = (dn && rt) ? p[(i + 1) * WW + (j + 1)] : 0.f;
            float gx = (tl - tr) + 2.f * (ml - mr) + (bl - br);
            float gy = (tl + 2.f * tc + tr) - (bl + 2.f * bc + br);
            mag += sqrtf(gx * gx + gy * gy + 1e-5f);
            sx += gx;
            sy += gy;
        }
    }
    const float R2D = (float)(180.0 / 3.14159);  // matches reference constant
    float ang = atan2f(sy, sx) * R2D + 180.0f;
    float ori = rintf(ang / 45.0f) * 45.0f;      // jnp.round = RNE

    grad_mag[gid] = mag;                          // re-read by NMS: keep RT
    grad_ori[gid] = ori;                          // re-read by NMS: keep RT
    __builtin_nontemporal_store((mag < 3.0f) ? 0.f : mag, &early_thr[gid]);
}

// ---------------------------------------------------------------------------
// Kernel C: non-max suppression + threshold. The 8-filter directional conv is
// just (center - directional neighbor), zero-padded. Both outputs are final
// (never re-read) -> non-temporal stores.
// ---------------------------------------------------------------------------
__global__ __launch_bounds__(256) void canny_nms(
    const float* __restrict__ grad_mag, const float* __restrict__ grad_ori,
    float* __restrict__ thin, float* __restrict__ thr)
{
    const int gid = blockIdx.x * 256 + threadIdx.x;
    const int b   = gid >> 18;
    const int pix = gid & (PIX - 1);
    const int i = pix >> 9, j = pix & 511;
    const float* __restrict__ m = grad_mag + b * PIX;

    const float mc = m[pix];
    const float o  = grad_ori[gid];
    const int idx  = ((int)(o / 45.0f)) & 7;       // o is an exact multiple of 45
    const int idn  = (idx + 4) & 7;

    const int di[8] = {0, 1, 1,  1,  0, -1, -1, -1};
    const int dj[8] = {1, 1, 0, -1, -1, -1,  0,  1};

    int rp = i + di[idx], cp = j + dj[idx];
    int rn = i + di[idn], cn = j + dj[idn];
    float np_ = (rp >= 0 && rp < HH && cp >= 0 && cp < WW) ? m[rp * WW + cp] : 0.f;
    float nn_ = (rn >= 0 && rn < HH && cn >= 0 && cn < WW) ? m[rn * WW + cn] : 0.f;

    float selp = mc - np_;
    float seln = mc - nn_;
    float t = (fminf(selp, seln) > 0.f) ? mc : 0.f;
    __builtin_nontemporal_store(t, &thin[gid]);
    __builtin_nontemporal_store((t < 3.0f) ? 0.f : t, &thr[gid]);
}

// ---------------------------------------------------------------------------
extern "C" void kernel_launch(void* const* d_in, const int* in_sizes, int n_in,
                              void* d_out, int out_size, void* d_ws, size_t ws_size,
                              hipStream_t stream)
{
    const float* img     = (const float*)d_in[0];
    const float* gauss_h = (const float*)d_in[1];
    // d_in[2..5]: gauss_v / sobel_h / sobel_v / dir_w (fixed stencils, folded in)

    float* out     = (float*)d_out;
    float* blurred = out;                                   // [1,32,3,512,512]
    float* gmag    = blurred + (size_t)IMG_N * PIX;         // [32,1,512,512]
    float* gori    = gmag    + (size_t)BATCH * PIX;
    float* thin    = gori    + (size_t)BATCH * PIX;
    float* thr     = thin    + (size_t)BATCH * PIX;
    float* early   = thr     + (size_t)BATCH * PIX;

    // 96 images * 32*32 tiles = 98304 tiles, 4 waves (tiles) per block
    canny_blur_wmma<<<98304 / 4, 128, 0, stream>>>(img, gauss_h, blurred);

    const int npix_blocks = (BATCH * PIX) / 256;            // 32768
    canny_grad<<<npix_blocks, 256, 0, stream>>>(blurred, gmag, gori, early);
    canny_nms<<<npix_blocks, 256, 0, stream>>>(gmag, gori, thin, thr);
}